// FourierSpectralBlock_60078002537067
// MI455X (gfx1250) — compile-verified
//
#include <hip/hip_runtime.h>
#include <math.h>

typedef __attribute__((ext_vector_type(2))) float v2f;
typedef __attribute__((ext_vector_type(8))) float v8f;

#define PI2 6.28318530717958647692f

static constexpr int IMG  = 96;
static constexpr int RW   = 49;              // 96/2+1
static constexpr int NB   = 2, NC = 32;
static constexpr int CR   = 24, RWC = 13;
static constexpr int SP   = CR * CR * RWC;   // 7488
static constexpr long long ROWS_W   = (long long)NB * NC * IMG * IMG;       // 589824
static constexpr long long NCOMPLEX = (long long)NB * NC * IMG * IMG * RW;  // 28901376
static constexpr long long VOL      = (long long)IMG * IMG * IMG;           // 884736
static constexpr long long TOT      = (long long)NB * NC * VOL;             // 56623104

__device__ __forceinline__ v8f wmma4(v2f a, v2f b, v8f c) {
  // D = A(16x4 f32) * B(4x16 f32) + C(16x16 f32)
  return __builtin_amdgcn_wmma_f32_16x16x4_f32(false, a, false, b, (short)0, c,
                                               false, false);
}

// ---- sortable-uint encoding for float atomics (min/max) ----
__device__ __forceinline__ unsigned fkey(float f) {
  unsigned u = __float_as_uint(f);
  return (u & 0x80000000u) ? ~u : (u | 0x80000000u);
}
__device__ __forceinline__ float fval(unsigned k) {
  unsigned u = (k & 0x80000000u) ? (k ^ 0x80000000u) : ~k;
  return __uint_as_float(u);
}

// =====================================================================
// 1) rfft along W (contiguous, 96 real -> 49 complex), DFT-as-GEMM.
//    4 waves/WG share one LDS tile of 16 rows x 96.
// =====================================================================
__global__ void k_rfft_w(const float* __restrict__ x, float2* __restrict__ out) {
  __shared__ float sk[16 * 96];           // [row][k]
  const int lane = threadIdx.x & 31, wave = threadIdx.x >> 5;
  const int half = lane >> 4, l16 = lane & 15;
  const int rbase = blockIdx.x * 16;

  for (int t = threadIdx.x; t < 16 * 96; t += blockDim.x) {
    int n = t / 96, k = t - n * 96;
    sk[t] = x[(size_t)(rbase + n) * 96 + k];
  }
  __syncthreads();

  const int mbase = wave * 16;
  const int m = mbase + l16;
  v8f accC = {}, accS = {};
  for (int ks = 0; ks < 24; ++ks) {
    const int k0 = ks * 4 + 2 * half;
    int p0 = (m * k0) % 96, p1 = (m * (k0 + 1)) % 96;
    float s0, c0, s1, c1;
    __sincosf(PI2 * (float)p0 * (1.0f / 96.0f), &s0, &c0);
    __sincosf(PI2 * (float)p1 * (1.0f / 96.0f), &s1, &c1);
    v2f Ac = {c0, c1}, As = {s0, s1};
    v2f Bx = {sk[l16 * 96 + k0], sk[l16 * 96 + k0 + 1]};
    accC = wmma4(Ac, Bx, accC);             // Fr = sum cos * x
    accS = wmma4(As, Bx, accS);             // Fi = -sum sin * x
  }
  for (int r = 0; r < 8; ++r) {
    int M = mbase + r + 8 * half;
    if (M < RW) {
      float2 o; o.x = accC[r]; o.y = -accS[r];
      out[(size_t)(rbase + l16) * RW + M] = o;
    }
  }
}

// =====================================================================
// 2) Generic complex DFT along a strided axis (layout [outer, Kbins, inner]).
//    freq(bin k) = k + freqOff ; phase = sgn * 2*pi * m * freq / Nout.
//    Nout/16 waves per WG, each produces a 16(m) x 16(inner) tile.
//    Kbins must be a multiple of 4 (96 / 32 / 24).
// =====================================================================
__global__ void k_cdft(const float2* __restrict__ in, float2* __restrict__ out,
                       int inner, int Kbins, int freqOff, int Nout, float sgn) {
  __shared__ float2 sk[96 * 16];          // [k][i]
  const int o = blockIdx.y;
  const int ibase = blockIdx.x * 16;
  const int lane = threadIdx.x & 31, wave = threadIdx.x >> 5;
  const int half = lane >> 4, l16 = lane & 15;

  for (int t = threadIdx.x; t < Kbins * 16; t += blockDim.x) {
    int k = t >> 4, i = t & 15;
    float2 v; v.x = 0.f; v.y = 0.f;
    if (ibase + i < inner) v = in[((size_t)o * Kbins + k) * inner + ibase + i];
    sk[t] = v;
  }
  __syncthreads();

  const int mbase = wave * 16;
  const int m = mbase + l16;
  const float w = sgn * PI2 / (float)Nout;
  v8f aCR = {}, aSI = {}, aCI = {}, aSR = {};
  for (int ks = 0; ks < (Kbins >> 2); ++ks) {
    const int k0 = ks * 4 + 2 * half;
    int p0 = (m * (k0 + freqOff)) % Nout;     if (p0 < 0) p0 += Nout;
    int p1 = (m * (k0 + 1 + freqOff)) % Nout; if (p1 < 0) p1 += Nout;
    float s0, c0, s1, c1;
    __sincosf(w * (float)p0, &s0, &c0);
    __sincosf(w * (float)p1, &s1, &c1);
    v2f Ac = {c0, c1}, As = {s0, s1};
    float2 x0 = sk[k0 * 16 + l16];
    float2 x1 = sk[(k0 + 1) * 16 + l16];
    v2f Bxr = {x0.x, x1.x}, Bxi = {x0.y, x1.y};
    aCR = wmma4(Ac, Bxr, aCR);
    aSI = wmma4(As, Bxi, aSI);
    aCI = wmma4(Ac, Bxi, aCI);
    aSR = wmma4(As, Bxr, aSR);
  }
  for (int r = 0; r < 8; ++r) {
    int M = mbase + r + 8 * half;
    int Nn = ibase + l16;
    if (Nn < inner) {
      float2 ov; ov.x = aCR[r] - aSI[r]; ov.y = aSR[r] + aCI[r];
      out[((size_t)o * Nout + M) * inner + Nn] = ov;
    }
  }
}

// =====================================================================
// 3) Size-2 butterfly over the batch axis (forward FFT of size 2).
// =====================================================================
__global__ void k_bfly(const float2* __restrict__ in, float2* __restrict__ out,
                       size_t h) {
  size_t i = (size_t)blockIdx.x * 256 + threadIdx.x;
  if (i >= h) return;
  float2 a = in[i], c = in[i + h], p, q;
  p.x = a.x + c.x; p.y = a.y + c.y;
  q.x = a.x - c.x; q.y = a.y - c.y;
  out[i] = p; out[i + h] = q;
}

// =====================================================================
// 4) Global min/max of real & imag parts (sortable-uint atomics).
// =====================================================================
__global__ void k_minmax_init(unsigned* mm) {
  mm[0] = 0xFFFFFFFFu; mm[1] = 0u; mm[2] = 0xFFFFFFFFu; mm[3] = 0u;
}
__global__ void k_minmax(const float2* __restrict__ f, size_t n, unsigned* mm) {
  __shared__ unsigned s0[256], s1[256], s2[256], s3[256];
  unsigned rmn = 0xFFFFFFFFu, rmx = 0u, imn = 0xFFFFFFFFu, imx = 0u;
  for (size_t i = (size_t)blockIdx.x * 256 + threadIdx.x; i < n;
       i += (size_t)gridDim.x * 256) {
    float2 v = f[i];
    unsigned kr = fkey(v.x), ki = fkey(v.y);
    rmn = kr < rmn ? kr : rmn; rmx = kr > rmx ? kr : rmx;
    imn = ki < imn ? ki : imn; imx = ki > imx ? ki : imx;
  }
  s0[threadIdx.x] = rmn; s1[threadIdx.x] = rmx;
  s2[threadIdx.x] = imn; s3[threadIdx.x] = imx;
  __syncthreads();
  for (int off = 128; off > 0; off >>= 1) {
    if ((int)threadIdx.x < off) {
      unsigned a;
      a = s0[threadIdx.x + off]; if (a < s0[threadIdx.x]) s0[threadIdx.x] = a;
      a = s1[threadIdx.x + off]; if (a > s1[threadIdx.x]) s1[threadIdx.x] = a;
      a = s2[threadIdx.x + off]; if (a < s2[threadIdx.x]) s2[threadIdx.x] = a;
      a = s3[threadIdx.x + off]; if (a > s3[threadIdx.x]) s3[threadIdx.x] = a;
    }
    __syncthreads();
  }
  if (threadIdx.x == 0) {
    atomicMin(&mm[0], s0[0]); atomicMax(&mm[1], s1[0]);
    atomicMin(&mm[2], s2[0]); atomicMax(&mm[3], s3[0]);
  }
}

// =====================================================================
// 5) fftshift + center-crop + [0,1] normalize -> y_crop [2,64,7488].
//    Shifted (b,c) maps to natural bins kb=b^1, kc=(c+16)&31.
// =====================================================================
__global__ void k_crop(const float2* __restrict__ spec,
                       const unsigned* __restrict__ mm,
                       float* __restrict__ yc) {
  int idx = blockIdx.x * 256 + threadIdx.x;
  if (idx >= NB * 64 * SP) return;
  int s = idx % SP;
  int ch = (idx / SP) & 63;
  int b = idx / (SP * 64);
  int jw = s % RWC, jh = (s / RWC) % CR, jd = s / (RWC * CR);
  int kd = (jd + 84) % 96, kh = (jh + 84) % 96, kw = (jw + 43) % 49;
  int kc = ((ch & 31) + 16) & 31, kb = b ^ 1;
  float2 v = spec[(((size_t)(kb * NC + kc) * IMG + kd) * IMG + kh) * RW + kw];
  float rmin = fval(mm[0]), rmax = fval(mm[1]);
  float imin = fval(mm[2]), imax = fval(mm[3]);
  yc[idx] = (ch < 32) ? (v.x - rmin) / (rmax - rmin)
                      : (v.y - imin) / (imax - imin);
}

// =====================================================================
// 6) Gate 1x1 conv GEMM: g[b,o,s] = sum_i gw[o,i]*y[b,i,s] + gb[o]  (WMMA)
// =====================================================================
__global__ void k_gate_gemm(const float* __restrict__ y, const float* __restrict__ gw,
                            const float* __restrict__ gb, float* __restrict__ g) {
  __shared__ float sk[64 * 16];           // [i][s]
  const int b = blockIdx.y;
  const int sbase = blockIdx.x * 16;
  const int lane = threadIdx.x & 31, wave = threadIdx.x >> 5;
  const int half = lane >> 4, l16 = lane & 15;

  for (int t = threadIdx.x; t < 64 * 16; t += blockDim.x) {
    int i = t >> 4, s = t & 15;
    sk[t] = y[((size_t)b * 64 + i) * SP + sbase + s];
  }
  __syncthreads();

  const int mbase = wave * 16;
  const int m = mbase + l16;
  v8f acc = {};
  for (int ks = 0; ks < 16; ++ks) {
    int k0 = ks * 4 + 2 * half;
    v2f A = {gw[m * 64 + k0], gw[m * 64 + k0 + 1]};
    v2f Bv = {sk[k0 * 16 + l16], sk[(k0 + 1) * 16 + l16]};
    acc = wmma4(A, Bv, acc);
  }
  for (int r = 0; r < 8; ++r) {
    int M = mbase + r + 8 * half;
    g[((size_t)b * 32 + M) * SP + sbase + l16] = acc[r] + gb[M];
  }
}

// =====================================================================
// 7) Per-group mean / rstd (instance norm stats), group = contiguous L elems.
// =====================================================================
__global__ void k_stats(const float* __restrict__ in, float* __restrict__ mean,
                        float* __restrict__ rstd, int L) {
  __shared__ float s1[256], s2[256];
  const float* p = in + (size_t)blockIdx.x * L;
  float a = 0.f, b = 0.f;
  for (int i = threadIdx.x; i < L; i += 256) { float v = p[i]; a += v; b += v * v; }
  s1[threadIdx.x] = a; s2[threadIdx.x] = b;
  __syncthreads();
  for (int off = 128; off > 0; off >>= 1) {
    if ((int)threadIdx.x < off) {
      s1[threadIdx.x] += s1[threadIdx.x + off];
      s2[threadIdx.x] += s2[threadIdx.x + off];
    }
    __syncthreads();
  }
  if (threadIdx.x == 0) {
    float m = s1[0] / (float)L;
    float var = s2[0] / (float)L - m * m;
    mean[blockIdx.x] = m;
    rstd[blockIdx.x] = rsqrtf(var + 1e-5f);
  }
}

__global__ void k_gate_sig(const float* __restrict__ g, const float* __restrict__ gm,
                           const float* __restrict__ gr, float* __restrict__ filt) {
  int idx = blockIdx.x * 256 + threadIdx.x;
  if (idx >= NB * NC * SP) return;
  int grp = idx / SP;
  float z = (g[idx] - gm[grp]) * gr[grp];
  filt[idx] = 1.0f / (1.0f + expf(-z));
}

// =====================================================================
// 8) Frequency layer: instnorm -> ew_W*x+ew_B -> depthwise scale -> denorm
//    -> gate; write complex cropped spectrum at NATURAL (kb,kc) bins.
// =====================================================================
__global__ void k_freq(const float* __restrict__ yc, const float* __restrict__ ycm,
                       const float* __restrict__ ycr, const float* __restrict__ ewW,
                       const float* __restrict__ ewB, const float* __restrict__ fw,
                       const float* __restrict__ fb, const float* __restrict__ filt,
                       const unsigned* __restrict__ mm, float2* __restrict__ spec) {
  int idx = blockIdx.x * 256 + threadIdx.x;
  if (idx >= NB * 64 * SP) return;
  int s = idx % SP;
  int ch = (idx / SP) & 63;
  int b = idx / (SP * 64);
  int grp = b * 64 + ch;
  float t = (yc[idx] - ycm[grp]) * ycr[grp];
  t = t * ewW[ch * SP + s] + ewB[ch * SP + s];
  t = t * fw[ch] + fb[ch];
  float f = filt[((size_t)b * 32 + (ch & 31)) * SP + s];
  int kb = b ^ 1, kc = ((ch & 31) + 16) & 31;   // ifftshift of b,c axes
  float* dst = (float*)&spec[((size_t)kb * 32 + kc) * SP + s];
  if (ch < 32) dst[0] = (t * (fval(mm[1]) - fval(mm[0])) + fval(mm[0])) * f;
  else         dst[1] = (t * (fval(mm[3]) - fval(mm[2])) + fval(mm[2])) * f;
}

// =====================================================================
// 9) Hermitian irfft along W: 13 nonzero bins -> 96 reals. WMMA, K padded
//    to 16. Folds the irfftn 1/96^3 normalization.
// =====================================================================
__global__ void k_irfft_w(const float2* __restrict__ in, float* __restrict__ rec) {
  __shared__ float2 sk[16 * 16];          // [row][j]
  const int lane = threadIdx.x & 31, wave = threadIdx.x >> 5;
  const int half = lane >> 4, l16 = lane & 15;
  const int rbase = blockIdx.x * 16;

  for (int t = threadIdx.x; t < 16 * 16; t += blockDim.x) {
    int n = t >> 4, j = t & 15;
    float2 v; v.x = 0.f; v.y = 0.f;
    if (j < RWC) v = in[(size_t)(rbase + n) * RWC + j];
    sk[t] = v;
  }
  __syncthreads();

  const int mbase = wave * 16;
  const int m = mbase + l16;              // output sample n in [0,96)
  const float scale = 1.0f / (96.0f * 96.0f * 96.0f);
  v8f acc = {};
  for (int ks = 0; ks < 4; ++ks) {
    const int j0 = ks * 4 + 2 * half;
    float cr[2] = {0.f, 0.f}, ci[2] = {0.f, 0.f};
#pragma unroll
    for (int q = 0; q < 2; ++q) {
      int j = j0 + q;
      if (j < RWC) {
        int kb = (43 + j) % 49;
        float wgt = (kb == 0 || kb == 48) ? 1.0f : 2.0f;
        int p = (kb * m) % 96;
        float s, c;
        __sincosf(PI2 * (float)p * (1.0f / 96.0f), &s, &c);
        cr[q] = wgt * c * scale;
        ci[q] = -wgt * s * scale;
      }
    }
    v2f Ar = {cr[0], cr[1]}, Ai = {ci[0], ci[1]};
    float2 x0 = sk[l16 * 16 + j0];
    float2 x1 = sk[l16 * 16 + j0 + 1];
    v2f Bxr = {x0.x, x1.x}, Bxi = {x0.y, x1.y};
    acc = wmma4(Ar, Bxr, acc);
    acc = wmma4(Ai, Bxi, acc);
  }
  for (int r = 0; r < 8; ++r) {
    int M = mbase + r + 8 * half;
    rec[(size_t)(rbase + l16) * 96 + M] = acc[r];
  }
}

// =====================================================================
// 10) rec += spatial path (depthwise scale+bias of x); then GELU(instnorm).
// =====================================================================
__global__ void k_add_sp(float* __restrict__ rec, const float* __restrict__ x,
                         const float* __restrict__ sw, const float* __restrict__ sb) {
  size_t i = (size_t)blockIdx.x * 256 + threadIdx.x;
  if (i >= (size_t)TOT) return;
  int c = (int)((i / (size_t)VOL) & 31);
  rec[i] += x[i] * sw[c] + sb[c];
}

__global__ void k_gelu(const float* __restrict__ rec, const float* __restrict__ fm,
                       const float* __restrict__ fr, float* __restrict__ out) {
  size_t i = (size_t)blockIdx.x * 256 + threadIdx.x;
  if (i >= (size_t)TOT) return;
  int grp = (int)(i / (size_t)VOL);
  float t = (rec[i] - fm[grp]) * fr[grp];
  out[i] = 0.5f * t * (1.0f + erff(t * 0.70710678118654752f));
}

// =====================================================================
extern "C" void kernel_launch(void* const* d_in, const int* in_sizes, int n_in,
                              void* d_out, int out_size, void* d_ws, size_t ws_size,
                              hipStream_t stream) {
  (void)in_sizes; (void)n_in; (void)out_size; (void)ws_size;
  const float* x   = (const float*)d_in[0];
  const float* sw  = (const float*)d_in[1];
  const float* sb  = (const float*)d_in[2];
  const float* ewW = (const float*)d_in[3];
  const float* ewB = (const float*)d_in[4];
  const float* fw  = (const float*)d_in[5];
  const float* fb  = (const float*)d_in[6];
  const float* gw  = (const float*)d_in[7];
  const float* gb  = (const float*)d_in[8];
  float* out = (float*)d_out;

  char* ws = (char*)d_ws;
  size_t off = 0;
  auto alloc = [&](size_t bytes) -> void* {
    void* p = (void*)(ws + off);
    off += (bytes + 255) & ~(size_t)255;
    return p;
  };
  float2*   bufA = (float2*)alloc((size_t)NCOMPLEX * 8);
  float2*   bufB = (float2*)alloc((size_t)NCOMPLEX * 8);
  float*    rec  = (float*) alloc((size_t)TOT * 4);
  float*    yc   = (float*) alloc((size_t)NB * 64 * SP * 4);
  float*    g    = (float*) alloc((size_t)NB * 32 * SP * 4);
  float*    filt = (float*) alloc((size_t)NB * 32 * SP * 4);
  float2*   spec = (float2*)alloc((size_t)NB * 32 * SP * 8);
  float*    st   = (float*) alloc(512 * 4);
  unsigned* mm   = (unsigned*)alloc(16);
  float *ycm = st, *ycr = st + 128, *gm = st + 256, *gr = st + 320,
        *fm = st + 384, *fr = st + 448;

  // ---- forward 5D rfftn (natural bin order; shifts handled as index maps)
  k_minmax_init<<<1, 1, 0, stream>>>(mm);
  k_rfft_w<<<(int)(ROWS_W / 16), 128, 0, stream>>>(x, bufA);
  k_cdft<<<dim3(4, NB * NC * IMG), 192, 0, stream>>>(bufA, bufB, RW, 96, 0, 96, -1.f);   // H
  k_cdft<<<dim3(IMG * RW / 16, NB * NC), 192, 0, stream>>>(bufB, bufA, IMG * RW, 96, 0, 96, -1.f); // D
  k_cdft<<<dim3((int)(IMG * IMG * RW / 16), NB), 64, 0, stream>>>(bufA, bufB,
        IMG * IMG * RW, 32, 0, 32, -1.f);                                                 // C
  k_bfly<<<(int)(NCOMPLEX / 2 / 256), 256, 0, stream>>>(bufB, bufA, (size_t)(NCOMPLEX / 2)); // B

  // ---- global min/max, crop+normalize, gate, frequency layer
  k_minmax<<<2048, 256, 0, stream>>>(bufA, (size_t)NCOMPLEX, mm);
  k_crop<<<NB * 64 * SP / 256, 256, 0, stream>>>(bufA, mm, yc);
  k_stats<<<NB * 64, 256, 0, stream>>>(yc, ycm, ycr, SP);
  k_gate_gemm<<<dim3(SP / 16, NB), 64, 0, stream>>>(yc, gw, gb, g);
  k_stats<<<NB * 32, 256, 0, stream>>>(g, gm, gr, SP);
  k_gate_sig<<<NB * 32 * SP / 256, 256, 0, stream>>>(g, gm, gr, filt);
  k_freq<<<NB * 64 * SP / 256, 256, 0, stream>>>(yc, ycm, ycr, ewW, ewB, fw, fb,
                                                 filt, mm, spec);

  // ---- pruned inverse over spatial axes only (24/24/13 nonzero bins)
  k_cdft<<<dim3(20, NB * NC), 192, 0, stream>>>(spec, bufB, CR * RWC, 24, -12, 96, 1.f);  // D^-1
  k_cdft<<<dim3(1, NB * NC * IMG), 192, 0, stream>>>(bufB, bufA, RWC, 24, -12, 96, 1.f);  // H^-1
  k_irfft_w<<<(int)(ROWS_W / 16), 192, 0, stream>>>(bufA, rec);                            // W^-1

  // ---- combine spatial path, instance norm, GELU
  k_add_sp<<<(int)(TOT / 256), 256, 0, stream>>>(rec, x, sw, sb);
  k_stats<<<NB * NC, 256, 0, stream>>>(rec, fm, fr, (int)VOL);
  k_gelu<<<(int)(TOT / 256), 256, 0, stream>>>(rec, fm, fr, out);
}